// GNN_6476810682405
// MI455X (gfx1250) — compile-verified
//
#include <hip/hip_runtime.h>

#define N_NODES 50000
#define N_EDGES 800000
#define D 256
#define N_LAYERS 3
#define N_GRAPHS 64
#define LN_EPS 1e-5f

typedef __attribute__((ext_vector_type(2))) float v2f;
typedef __attribute__((ext_vector_type(8))) float v8f;

// ---------------------------------------------------------------- prepass ---

__global__ void gnn_zero_u32(unsigned* __restrict__ p, int n) {
    int i = blockIdx.x * blockDim.x + threadIdx.x;
    if (i < n) p[i] = 0u;
}

__global__ void gnn_hist(const int* __restrict__ dst, unsigned* __restrict__ cnt) {
    int e = blockIdx.x * blockDim.x + threadIdx.x;
    if (e < N_EDGES) atomicAdd(&cnt[dst[e]], 1u);
}

__global__ void gnn_dinv(const unsigned* __restrict__ cnt, float* __restrict__ dinv) {
    int i = blockIdx.x * blockDim.x + threadIdx.x;
    if (i < N_NODES) dinv[i] = rsqrtf((float)cnt[i] + 1.0f);  // +1 self-loop
}

// Single-block exclusive scan over 50000 counts -> CSR row offsets + cursor copy.
__global__ void gnn_scan(const unsigned* __restrict__ cnt,
                         unsigned* __restrict__ row_off,
                         unsigned* __restrict__ cursor) {
    __shared__ unsigned sh[1024];
    const int t = threadIdx.x;
    const int chunk = (N_NODES + 1023) / 1024;
    int start = t * chunk; if (start > N_NODES) start = N_NODES;
    int end = start + chunk; if (end > N_NODES) end = N_NODES;
    unsigned s = 0;
    for (int i = start; i < end; ++i) s += cnt[i];
    sh[t] = s;
    __syncthreads();
    for (int off = 1; off < 1024; off <<= 1) {
        unsigned v = (t >= off) ? sh[t - off] : 0u;
        __syncthreads();
        sh[t] += v;
        __syncthreads();
    }
    unsigned run = (t == 0) ? 0u : sh[t - 1];
    for (int i = start; i < end; ++i) {
        row_off[i] = run;
        cursor[i]  = run;
        run += cnt[i];
    }
    if (t == 1023) row_off[N_NODES] = run;  // == N_EDGES
}

__global__ void gnn_csr_fill(const int* __restrict__ src, const int* __restrict__ dst,
                             unsigned* __restrict__ cursor, int* __restrict__ csr_src) {
    int e = blockIdx.x * blockDim.x + threadIdx.x;
    if (e < N_EDGES) {
        unsigned p = atomicAdd(&cursor[dst[e]], 1u);
        csr_src[p] = src[e];
    }
}

// Wt[l][n][k] = W[l][k][n]  (so the WMMA B operand loads as contiguous float2)
__global__ void gnn_transpose_w(const float* __restrict__ W, float* __restrict__ Wt) {
    int idx = blockIdx.x * blockDim.x + threadIdx.x;
    if (idx < N_LAYERS * D * D) {
        int l = idx / (D * D);
        int r = idx % (D * D);
        int k = r / D, n = r % D;
        Wt[(size_t)l * D * D + (size_t)n * D + k] = W[idx];
    }
}

// ------------------------------------------------------------- WMMA GEMM ---
// H[50000x256] = X[50000x256] @ W[256x256], fp32 via V_WMMA_F32_16X16X4_F32.
// Block = 128 threads (4 waves). Block owns a 16-row M tile; wave w owns the
// 16x64 output slab at columns [64w, 64w+64). 64 K-steps of 4.
//
// fp32 WMMA operand layout (ISA 7.12.2):
//   A 16x4:  lanes 0-15 -> M=lane,   VGPR{0,1} = K{0,1}; lanes 16-31 -> K{2,3}
//   B 4x16:  lanes 0-15 -> N=lane,   VGPR{0,1} = K{0,1}; lanes 16-31 -> K{2,3}
//   => per lane both operands read K = 4*kb + 2*(lane>>4) + {0,1}, i.e. one
//      aligned float2 from the row (A) / transposed-row (B).
//   C/D 16x16: VGPR v -> row = v + 8*(lane>>4), col = lane&15.
__global__ __launch_bounds__(128) void gnn_gemm_wmma(const float* __restrict__ X,
                                                     const float* __restrict__ Wt,
                                                     float* __restrict__ H) {
    const int wave = threadIdx.x >> 5;   // N-tile group 0..3
    const int lane = threadIdx.x & 31;
    const int half = lane >> 4;
    const int l16  = lane & 15;
    const int Mbase = blockIdx.x * 16;
    const int Nbase = wave * 64;
    const int row = Mbase + l16;

    const float2* __restrict__ A  = (const float2*)(X + (size_t)row * D);
    const float2* __restrict__ B0 = (const float2*)(Wt + (size_t)(Nbase +  0 + l16) * D);
    const float2* __restrict__ B1 = (const float2*)(Wt + (size_t)(Nbase + 16 + l16) * D);
    const float2* __restrict__ B2 = (const float2*)(Wt + (size_t)(Nbase + 32 + l16) * D);
    const float2* __restrict__ B3 = (const float2*)(Wt + (size_t)(Nbase + 48 + l16) * D);

    v8f acc0 = {}, acc1 = {}, acc2 = {}, acc3 = {};

#pragma unroll 4
    for (int kb = 0; kb < 64; ++kb) {
        const int ki = kb * 2 + half;          // float2 index: K = 4*kb + 2*half
        float2 af = A[ki];
        v2f a = {af.x, af.y};
        float2 f0 = B0[ki], f1 = B1[ki], f2 = B2[ki], f3 = B3[ki];
        v2f b0 = {f0.x, f0.y}, b1 = {f1.x, f1.y}, b2 = {f2.x, f2.y}, b3 = {f3.x, f3.y};
        acc0 = __builtin_amdgcn_wmma_f32_16x16x4_f32(false, a, false, b0, (short)0, acc0, false, false);
        acc1 = __builtin_amdgcn_wmma_f32_16x16x4_f32(false, a, false, b1, (short)0, acc1, false, false);
        acc2 = __builtin_amdgcn_wmma_f32_16x16x4_f32(false, a, false, b2, (short)0, acc2, false, false);
        acc3 = __builtin_amdgcn_wmma_f32_16x16x4_f32(false, a, false, b3, (short)0, acc3, false, false);
    }

#pragma unroll
    for (int v = 0; v < 8; ++v) {
        const int r = Mbase + half * 8 + v;
        float* __restrict__ hp = H + (size_t)r * D + l16;
        hp[Nbase +  0] = acc0[v];
        hp[Nbase + 16] = acc1[v];
        hp[Nbase + 32] = acc2[v];
        hp[Nbase + 48] = acc3[v];
    }
}

// ------------------------------------- fused aggregate + bias + LN + ReLU ---

__device__ __forceinline__ void fma4(float4& a, float c, const float4& v) {
    a.x = fmaf(c, v.x, a.x); a.y = fmaf(c, v.y, a.y);
    a.z = fmaf(c, v.z, a.z); a.w = fmaf(c, v.w, a.w);
}

// One wave per node. Lane L owns dims {4L..4L+3} and {128+4L..128+4L+3}.
__global__ __launch_bounds__(256) void gnn_agg_ln(const float* __restrict__ H,
                                                  const float* __restrict__ dinv,
                                                  const unsigned* __restrict__ row_off,
                                                  const int* __restrict__ csr_src,
                                                  const float* __restrict__ bias,
                                                  const float* __restrict__ gamma,
                                                  const float* __restrict__ beta,
                                                  float* __restrict__ Xout) {
    const int wave = threadIdx.x >> 5;
    const int lane = threadIdx.x & 31;
    const int n = blockIdx.x * 8 + wave;
    if (n >= N_NODES) return;

    const float dn = dinv[n];
    const float4* __restrict__ hn = (const float4*)(H + (size_t)n * D);
    float4 a0 = hn[lane], a1 = hn[lane + 32];
    const float selfc = dn * dn;                       // self-loop term
    float4 acc0 = make_float4(a0.x * selfc, a0.y * selfc, a0.z * selfc, a0.w * selfc);
    float4 acc1 = make_float4(a1.x * selfc, a1.y * selfc, a1.z * selfc, a1.w * selfc);

    const unsigned e0 = row_off[n], e1 = row_off[n + 1];
    for (unsigned e = e0; e < e1; ++e) {
        const int s = csr_src[e];
        const float c = dinv[s] * dn;
        const float4* __restrict__ hs = (const float4*)(H + (size_t)s * D);
        float4 v0 = hs[lane], v1 = hs[lane + 32];
        fma4(acc0, c, v0);
        fma4(acc1, c, v1);
    }

    const float4* __restrict__ b4 = (const float4*)bias;
    float4 bb0 = b4[lane], bb1 = b4[lane + 32];
    acc0.x += bb0.x; acc0.y += bb0.y; acc0.z += bb0.z; acc0.w += bb0.w;
    acc1.x += bb1.x; acc1.y += bb1.y; acc1.z += bb1.z; acc1.w += bb1.w;

    // LayerNorm over 256 dims: wave32 butterfly reduction of sum & sum-sq.
    float s  = acc0.x + acc0.y + acc0.z + acc0.w + acc1.x + acc1.y + acc1.z + acc1.w;
    float sq = acc0.x * acc0.x + acc0.y * acc0.y + acc0.z * acc0.z + acc0.w * acc0.w +
               acc1.x * acc1.x + acc1.y * acc1.y + acc1.z * acc1.z + acc1.w * acc1.w;
#pragma unroll
    for (int off = 16; off >= 1; off >>= 1) {
        s  += __shfl_xor(s,  off, 32);
        sq += __shfl_xor(sq, off, 32);
    }
    const float mu  = s * (1.0f / 256.0f);
    const float var = sq * (1.0f / 256.0f) - mu * mu;
    const float inv = rsqrtf(var + LN_EPS);

    const float4* __restrict__ g4  = (const float4*)gamma;
    const float4* __restrict__ be4 = (const float4*)beta;
    float4 g0 = g4[lane], g1 = g4[lane + 32];
    float4 t0 = be4[lane], t1 = be4[lane + 32];

    float4 o0, o1;
    o0.x = fmaxf(0.0f, (acc0.x - mu) * inv * g0.x + t0.x);
    o0.y = fmaxf(0.0f, (acc0.y - mu) * inv * g0.y + t0.y);
    o0.z = fmaxf(0.0f, (acc0.z - mu) * inv * g0.z + t0.z);
    o0.w = fmaxf(0.0f, (acc0.w - mu) * inv * g0.w + t0.w);
    o1.x = fmaxf(0.0f, (acc1.x - mu) * inv * g1.x + t1.x);
    o1.y = fmaxf(0.0f, (acc1.y - mu) * inv * g1.y + t1.y);
    o1.z = fmaxf(0.0f, (acc1.z - mu) * inv * g1.z + t1.z);
    o1.w = fmaxf(0.0f, (acc1.w - mu) * inv * g1.w + t1.w);

    float4* __restrict__ xo = (float4*)(Xout + (size_t)n * D);
    xo[lane]      = o0;
    xo[lane + 32] = o1;
}

// --------------------------------------------------------------- pooling ---
// One block per graph (batch is sorted); deterministic sequential reduce.
__global__ void gnn_pool(const float* __restrict__ X, const int* __restrict__ batch,
                         float* __restrict__ out) {
    const int g = blockIdx.x;
    const int d = threadIdx.x;
    int lo = 0, hi = N_NODES;
    while (lo < hi) { int m = (lo + hi) >> 1; if (batch[m] < g)     lo = m + 1; else hi = m; }
    int a = lo, b = N_NODES;
    while (a < b)   { int m = (a + b) >> 1;   if (batch[m] < g + 1) a = m + 1;  else b = m; }
    const int count = a - lo;
    float s = 0.0f;
    for (int n = lo; n < a; ++n) s += X[(size_t)n * D + d];
    out[(size_t)g * D + d] = (count > 0) ? s / (float)count : 0.0f;
}

// ---------------------------------------------------------------- launch ---

extern "C" void kernel_launch(void* const* d_in, const int* in_sizes, int n_in,
                              void* d_out, int out_size, void* d_ws, size_t ws_size,
                              hipStream_t stream) {
    const float* x      = (const float*)d_in[0];
    const int*   ei     = (const int*)d_in[1];
    const int*   src    = ei;
    const int*   dst    = ei + N_EDGES;
    const int*   batch  = (const int*)d_in[2];
    const float* Ws     = (const float*)d_in[3];
    const float* bs     = (const float*)d_in[4];
    const float* gammas = (const float*)d_in[5];
    const float* betas  = (const float*)d_in[6];
    float* out = (float*)d_out;

    char* wp = (char*)d_ws;
    auto carve = [&](size_t bytes) -> char* {
        char* p = wp;
        wp += (bytes + 255) & ~(size_t)255;
        return p;
    };
    float*    bufH    = (float*)carve((size_t)N_NODES * D * sizeof(float));
    float*    bufX    = (float*)carve((size_t)N_NODES * D * sizeof(float));
    float*    Wt      = (float*)carve((size_t)N_LAYERS * D * D * sizeof(float));
    float*    dinv    = (float*)carve((size_t)N_NODES * sizeof(float));
    unsigned* cnt     = (unsigned*)carve((size_t)N_NODES * sizeof(unsigned));
    unsigned* row_off = (unsigned*)carve((size_t)(N_NODES + 1) * sizeof(unsigned));
    unsigned* cursor  = (unsigned*)carve((size_t)N_NODES * sizeof(unsigned));
    int*      csr     = (int*)carve((size_t)N_EDGES * sizeof(int));

    gnn_zero_u32<<<(N_NODES + 255) / 256, 256, 0, stream>>>(cnt, N_NODES);
    gnn_hist<<<(N_EDGES + 255) / 256, 256, 0, stream>>>(dst, cnt);
    gnn_dinv<<<(N_NODES + 255) / 256, 256, 0, stream>>>(cnt, dinv);
    gnn_scan<<<1, 1024, 0, stream>>>(cnt, row_off, cursor);
    gnn_csr_fill<<<(N_EDGES + 255) / 256, 256, 0, stream>>>(src, dst, cursor, csr);
    gnn_transpose_w<<<(N_LAYERS * D * D + 255) / 256, 256, 0, stream>>>(Ws, Wt);

    const float* xin = x;
    for (int l = 0; l < N_LAYERS; ++l) {
        gnn_gemm_wmma<<<N_NODES / 16, 128, 0, stream>>>(xin, Wt + (size_t)l * D * D, bufH);
        gnn_agg_ln<<<(N_NODES + 7) / 8, 256, 0, stream>>>(bufH, dinv, row_off, csr,
                                                          bs + l * D, gammas + l * D,
                                                          betas + l * D, bufX);
        xin = bufX;
    }
    gnn_pool<<<N_GRAPHS, D, 0, stream>>>(bufX, batch, out);
}